// CARAFE_21148418965577
// MI455X (gfx1250) — compile-verified
//
#include <hip/hip_runtime.h>
#include <math.h>

typedef float v2f __attribute__((ext_vector_type(2)));
typedef float v8f __attribute__((ext_vector_type(8)));

#define HW 4096   // 64*64 input pixels per (b, channel) plane
#define OHW 16384 // 128*128 output pixels per (b, channel) plane

// ---------------------------------------------------------------------------
// Kernel 0: one-shot transpose of enc weights into tap-major K order:
//   Wt[n][tap*64 + ci] = We[n][ci*9 + tap]      (tap = kh*3 + kw)
// 225 KB, L2-resident for the enc GEMM; makes B-fragment loads contiguous.
// ---------------------------------------------------------------------------
__global__ void __launch_bounds__(256) transpose_encw(
    const float* __restrict__ We, float* __restrict__ Wt)
{
  int idx = blockIdx.x * 256 + threadIdx.x;   // 100*576 = 57600 threads
  int n  = idx / 576;
  int k  = idx - n * 576;                     // k = ci*9 + tap
  int ci = k / 9;
  int r  = k - ci * 9;
  Wt[n * 576 + r * 64 + ci] = We[idx];
}

// ---------------------------------------------------------------------------
// Kernel 1: comp 1x1 conv (GEMM M=4096,N=64,K=256) + BN + SiLU, fp32 WMMA.
// One wave = one 16-row M-tile x ALL of N (4 tiles): the A fragment is loaded
// once and feeds 4 consecutive WMMAs (X read exactly once from HBM).
// A[m,k] = X[b,k,m] ; B[k,n] = comp_w[n*256 + k] (64 KB, L2-resident)
// ---------------------------------------------------------------------------
__global__ void __launch_bounds__(256) comp_gemm_silu(
    const float* __restrict__ X, const float* __restrict__ Wc,
    const float* __restrict__ Sc, const float* __restrict__ Bc,
    float* __restrict__ W1)
{
  int wave = (blockIdx.x * 256 + threadIdx.x) >> 5;   // 2048 waves total
  int lane = threadIdx.x & 31;
  int bat  = wave >> 8;             // 256 M-tiles per batch
  int m0   = (wave & 255) << 4;
  int half = lane >> 4;             // 0: K pair {k,k+1}; 1: {k+2,k+3}
  int lid  = lane & 15;

  const float* Xb = X  + (size_t)bat * 256 * HW + (m0 + lid);
  const float* Wn = Wc + (size_t)lid * 256;   // n = nt*16 + lid -> +nt*4096

  v8f acc[4] = {};
  for (int k0 = 0; k0 < 256; k0 += 4) {
    int ka = k0 + half * 2;
    v2f a;
    a.x = Xb[(size_t)ka * HW];
    a.y = Xb[(size_t)(ka + 1) * HW];
#pragma unroll
    for (int nt = 0; nt < 4; ++nt) {
      v2f bb;
      bb.x = Wn[nt * 4096 + ka];
      bb.y = Wn[nt * 4096 + ka + 1];
      acc[nt] = __builtin_amdgcn_wmma_f32_16x16x4_f32(false, a, false, bb,
                                                      (short)0, acc[nt],
                                                      false, false);
    }
  }

  // Epilogue: BN affine + SiLU (fast rcp), store W1[b,n,m]; M = m0+r+8*half
#pragma unroll
  for (int nt = 0; nt < 4; ++nt) {
    int N = nt * 16 + lid;
    float s = Sc[N], bia = Bc[N];
    float* Ob = W1 + ((size_t)bat * 64 + N) * HW + m0 + half * 8;
#pragma unroll
    for (int r = 0; r < 8; ++r) {
      float v = acc[nt][r] * s + bia;
      v = v * __builtin_amdgcn_rcpf(1.0f + __expf(-v));   // SiLU
      Ob[r] = v;
    }
  }
}

// ---------------------------------------------------------------------------
// Kernel 2: enc 3x3 conv (im2col GEMM M=4096,N=100->112,K=576) + BN.
// K is walked tap-major (k' = tap*64 + ci): the tap's coordinates, bounds
// check and row pointer are hoisted out of the inner 16 K-steps, which then
// just stride contiguous channel planes. One wave = one M-tile x ALL of N
// (7 tiles), so each im2col A fragment feeds 7 consecutive WMMAs.
// ---------------------------------------------------------------------------
__global__ void __launch_bounds__(256) enc_gemm_bn(
    const float* __restrict__ W1, const float* __restrict__ Wt,
    const float* __restrict__ Se, const float* __restrict__ Be,
    float* __restrict__ W2)
{
  int wave = (blockIdx.x * 256 + threadIdx.x) >> 5;   // 2048 waves total
  int lane = threadIdx.x & 31;
  int bat  = wave >> 8;
  int m0   = (wave & 255) << 4;
  int half = lane >> 4;
  int lid  = lane & 15;
  int m    = m0 + lid, y = m >> 6, x = m & 63;

  const float* Ab = W1 + (size_t)bat * 64 * HW;

  const float* WnT[7];
  float msk[7];
#pragma unroll
  for (int nt = 0; nt < 7; ++nt) {
    int N = nt * 16 + lid;
    msk[nt] = (N < 100) ? 1.0f : 0.0f;
    WnT[nt] = Wt + (size_t)((N < 100) ? N : 99) * 576;
  }

  v8f acc[7] = {};
#pragma unroll
  for (int tap = 0; tap < 9; ++tap) {
    int kh = tap / 3, kw = tap - (tap / 3) * 3;     // constants after unroll
    int iy = y + kh - 1, ix = x + kw - 1;
    bool ok = ((unsigned)iy < 64u) & ((unsigned)ix < 64u);
    // per-tap base pointer, with this lane-half's K offset folded in
    const float* Ap = Ab + (ok ? (iy * 64 + ix) : 0) + (size_t)(half * 2) * HW;
    int kb = tap * 64 + half * 2;
    for (int cb = 0; cb < 64; cb += 4) {
      v2f a;
      a.x = ok ? Ap[(size_t)cb * HW]       : 0.0f;
      a.y = ok ? Ap[(size_t)(cb + 1) * HW] : 0.0f;
      int ka = kb + cb;
#pragma unroll
      for (int nt = 0; nt < 7; ++nt) {
        v2f bb;
        bb.x = msk[nt] * WnT[nt][ka];
        bb.y = msk[nt] * WnT[nt][ka + 1];
        acc[nt] = __builtin_amdgcn_wmma_f32_16x16x4_f32(false, a, false, bb,
                                                        (short)0, acc[nt],
                                                        false, false);
      }
    }
  }

#pragma unroll
  for (int nt = 0; nt < 7; ++nt) {
    int N = nt * 16 + lid;
    if (N < 100) {
      float s = Se[N], bia = Be[N];
      float* Ob = W2 + ((size_t)bat * 100 + N) * HW + m0 + half * 8;
#pragma unroll
      for (int r = 0; r < 8; ++r)
        Ob[r] = acc[nt][r] * s + bia;
    }
  }
}

// ---------------------------------------------------------------------------
// Kernel 3: softmax over the 25 reassembly weights in enc-channel space.
// PixelShuffle maps output (y,x) -> enc channels {4t+s}, s=(y&1)*2+(x&1),
// at input pixel (y/2, x/2). In-place on W2; never materializes the shuffle.
// ---------------------------------------------------------------------------
__global__ void __launch_bounds__(256) softmax25(float* __restrict__ W2)
{
  int idx = blockIdx.x * 256 + threadIdx.x;   // 8*4*4096 = 131072 threads
  int pix = idx & 4095;
  int s   = (idx >> 12) & 3;
  int bat = idx >> 14;
  float* base = W2 + (size_t)bat * 100 * HW + pix;

  float v[25];
  float vmax = -3.4e38f;
#pragma unroll
  for (int t = 0; t < 25; ++t) {
    v[t] = base[(size_t)(t * 4 + s) * HW];
    vmax = fmaxf(vmax, v[t]);
  }
  float sum = 0.0f;
#pragma unroll
  for (int t = 0; t < 25; ++t) { v[t] = __expf(v[t] - vmax); sum += v[t]; }
  float inv = __builtin_amdgcn_rcpf(sum);
#pragma unroll
  for (int t = 0; t < 25; ++t)
    base[(size_t)(t * 4 + s) * HW] = v[t] * inv;
}

// ---------------------------------------------------------------------------
// Kernel 4: reassembly. Output pixels (y, 2j) and (y, 2j+1) share the SAME
// 5x5 X neighborhood (only the weight parity s differs), so each lane owns
// one input column ix0, computes both outputs, and stores them as one b64.
// Halves the gather load count; X column loads are fully coalesced.
// ---------------------------------------------------------------------------
__global__ void __launch_bounds__(256) carafe_reassemble(
    const float* __restrict__ X, const float* __restrict__ W2,
    float* __restrict__ Out)
{
  int bid    = blockIdx.x;          // bat*256 + y*2 + xt  (2048 blocks)
  int xt     = bid & 1;
  int y      = (bid >> 1) & 127;
  int bat    = bid >> 8;
  int waveid = threadIdx.x >> 5;
  int lane   = threadIdx.x & 31;
  int ix0    = (xt << 5) + lane;    // input column (0..63) owned by this lane
  int x0     = ix0 << 1;            // output columns x0, x0+1
  int iy0    = y >> 1;
  int s0     = (y & 1) << 1;        // weight-parity for even x; +1 for odd x

  // 2 x 25 per-pixel softmax weights -> registers (reused by 256 channels)
  const float* Wb = W2 + (size_t)bat * 100 * HW + iy0 * 64 + ix0;
  float wgt0[25], wgt1[25];
#pragma unroll
  for (int t = 0; t < 25; ++t) {
    wgt0[t] = Wb[(size_t)(t * 4 + s0)     * HW];
    wgt1[t] = Wb[(size_t)(t * 4 + s0 + 1) * HW];
  }

  int  colidx[5];
  bool colok[5];
#pragma unroll
  for (int q = 0; q < 5; ++q) {
    int ix = ix0 + q - 2;
    colok[q]  = (unsigned)ix < 64u;
    colidx[q] = colok[q] ? ix : 0;
  }

  const float* Xb = X + (size_t)bat * 256 * HW;
  float2* Ob = (float2*)(Out + (size_t)bat * 256 * OHW + y * 128 + x0);

  for (int c = waveid; c < 256; c += 8) {
    const float* Xc = Xb + (size_t)c * HW;
    if (c + 8 < 256)
      __builtin_prefetch(Xc + 8 * HW + iy0 * 64, 0, 1);  // global_prefetch_b8
    float a0 = 0.0f, a1 = 0.0f;
#pragma unroll
    for (int p = 0; p < 5; ++p) {
      int iy = iy0 + p - 2;
      if ((unsigned)iy < 64u) {
        const float* row = Xc + iy * 64;
#pragma unroll
        for (int q = 0; q < 5; ++q) {
          float val = colok[q] ? row[colidx[q]] : 0.0f;
          a0 = fmaf(wgt0[p * 5 + q], val, a0);
          a1 = fmaf(wgt1[p * 5 + q], val, a1);
        }
      }
    }
    float2 o; o.x = a0; o.y = a1;
    Ob[(size_t)c * (OHW / 2)] = o;   // global_store_b64, coalesced
  }
}

// ---------------------------------------------------------------------------
extern "C" void kernel_launch(void* const* d_in, const int* in_sizes, int n_in,
                              void* d_out, int out_size, void* d_ws, size_t ws_size,
                              hipStream_t stream)
{
  const float* X  = (const float*)d_in[0];
  const float* Wc = (const float*)d_in[1];
  const float* Sc = (const float*)d_in[2];
  const float* Bc = (const float*)d_in[3];
  const float* We = (const float*)d_in[4];
  const float* Se = (const float*)d_in[5];
  const float* Be = (const float*)d_in[6];
  float* Out = (float*)d_out;

  // Workspace: W1 = 8*64*4096 f (8 MB), W2 = 8*100*4096 f (13.1 MB),
  //            Wt = 100*576 f (0.23 MB)
  float* W1 = (float*)d_ws;
  float* W2 = W1 + (size_t)8 * 64 * HW;
  float* Wt = W2 + (size_t)8 * 100 * HW;

  transpose_encw   <<< 225, 256, 0, stream>>>(We, Wt);
  comp_gemm_silu   <<< 256, 256, 0, stream>>>(X, Wc, Sc, Bc, W1);
  enc_gemm_bn      <<< 256, 256, 0, stream>>>(W1, Wt, Se, Be, W2);
  softmax25        <<< 512, 256, 0, stream>>>(W2);
  carafe_reassemble<<<2048, 256, 0, stream>>>(X, W2, Out);
}